// ModelTorch_60533269069827
// MI455X (gfx1250) — compile-verified
//
#include <hip/hip_runtime.h>

// Ragged segment compaction for MI455X (gfx1250, wave32).
// out[new_start[b] + j] = src[old_start[b] + j]  for j < filter[b]
//   old_start[b] = excl_cumsum(accept_length)[b] + b
//   new_start[b] = excl_cumsum(filter)[b]
//
// 3-phase plan (memory-bound; ~190MB traffic -> ~8us floor @ 23.3 TB/s):
//   1) per-block (1024 segs) partial sums of (al, filter)
//   2) single-block exclusive scan of the block sums (in place, in d_ws)
//   3) fused: re-scan block locally + one wave32 per segment copies its slice
//      with non-temporal loads/stores (data touched exactly once).

#define THREADS 256
#define PER_THREAD 4
#define ELEMS_PER_BLOCK (THREADS * PER_THREAD)   // 1024 segments per block
#define WAVE 32                                   // gfx1250 is wave32-only

__global__ void k_partial_sums(const int* __restrict__ al, const int* __restrict__ f,
                               int B, int* __restrict__ blkA, int* __restrict__ blkF) {
    __shared__ int sA[THREADS];
    __shared__ int sF[THREADS];
    const int tid  = threadIdx.x;
    const int base = blockIdx.x * ELEMS_PER_BLOCK;
    int a = 0, fl = 0;
#pragma unroll
    for (int k = 0; k < PER_THREAD; ++k) {
        int i = base + tid + k * THREADS;
        if (i < B) { a += al[i]; fl += f[i]; }
    }
    sA[tid] = a; sF[tid] = fl;
    __syncthreads();
    for (int off = THREADS / 2; off > 0; off >>= 1) {
        if (tid < off) { sA[tid] += sA[tid + off]; sF[tid] += sF[tid + off]; }
        __syncthreads();
    }
    if (tid == 0) { blkA[blockIdx.x] = sA[0]; blkF[blockIdx.x] = sF[0]; }
}

// Single block of 1024 threads: exclusive scan of NB block sums (in place).
// Handles NB > 1024 via sequential chunks with a carry.
__global__ void k_scan_sums(int* __restrict__ blkA, int* __restrict__ blkF, int NB) {
    __shared__ int sA[1024];
    __shared__ int sF[1024];
    const int tid = threadIdx.x;
    int carryA = 0, carryF = 0;
    for (int base = 0; base < NB; base += 1024) {
        int i = base + tid;
        int a  = (i < NB) ? blkA[i] : 0;
        int fl = (i < NB) ? blkF[i] : 0;
        sA[tid] = a; sF[tid] = fl;
        __syncthreads();
        for (int off = 1; off < 1024; off <<= 1) {
            int va = (tid >= off) ? sA[tid - off] : 0;
            int vf = (tid >= off) ? sF[tid - off] : 0;
            __syncthreads();
            sA[tid] += va; sF[tid] += vf;
            __syncthreads();
        }
        if (i < NB) {
            blkA[i] = carryA + sA[tid] - a;   // exclusive
            blkF[i] = carryF + sF[tid] - fl;
        }
        int tA = sA[1023], tF = sF[1023];     // chunk totals (tail padded w/ 0)
        __syncthreads();                       // protect LDS before next chunk
        carryA += tA; carryF += tF;
    }
}

__global__ void k_scan_copy(const float* __restrict__ src,
                            const int* __restrict__ al, const int* __restrict__ f,
                            int B,
                            const int* __restrict__ blkA, const int* __restrict__ blkF,
                            float* __restrict__ out) {
    __shared__ int scanA[THREADS];
    __shared__ int scanF[THREADS];
    __shared__ int sOld[ELEMS_PER_BLOCK];
    __shared__ int sNew[ELEMS_PER_BLOCK];
    __shared__ int sLen[ELEMS_PER_BLOCK];

    const int tid  = threadIdx.x;
    const int base = blockIdx.x * ELEMS_PER_BLOCK;
    const int g0   = base + tid * PER_THREAD;

    int aLoc[PER_THREAD], fLoc[PER_THREAD];
#pragma unroll
    for (int k = 0; k < PER_THREAD; ++k) {
        int g = g0 + k;
        aLoc[k] = (g < B) ? al[g] : 0;
        fLoc[k] = (g < B) ? f[g]  : 0;
    }
    int sumA = 0, sumF = 0;
#pragma unroll
    for (int k = 0; k < PER_THREAD; ++k) { sumA += aLoc[k]; sumF += fLoc[k]; }

    scanA[tid] = sumA; scanF[tid] = sumF;
    __syncthreads();
    for (int off = 1; off < THREADS; off <<= 1) {
        int va = (tid >= off) ? scanA[tid - off] : 0;
        int vf = (tid >= off) ? scanF[tid - off] : 0;
        __syncthreads();
        scanA[tid] += va; scanF[tid] += vf;
        __syncthreads();
    }
    int exclA = scanA[tid] - sumA + blkA[blockIdx.x];  // sum(al[0..g0))
    int exclF = scanF[tid] - sumF + blkF[blockIdx.x];  // sum(f [0..g0))

#pragma unroll
    for (int k = 0; k < PER_THREAD; ++k) {
        int g = g0 + k;
        if (g < B) {
            int li = tid * PER_THREAD + k;
            sOld[li] = exclA + g;      // excl_cumsum(al)[g] + g
            sNew[li] = exclF;          // excl_cumsum(f)[g]
            sLen[li] = fLoc[k];
        }
        exclA += aLoc[k];
        exclF += fLoc[k];
    }
    __syncthreads();

    // Copy phase: one wave32 per segment, round-robin over the block's waves.
    int nSeg = B - base;
    if (nSeg > ELEMS_PER_BLOCK) nSeg = ELEMS_PER_BLOCK;
    const int wave   = tid / WAVE;
    const int lane   = tid % WAVE;
    const int nWaves = THREADS / WAVE;
    for (int s = wave; s < nSeg; s += nWaves) {
        const int o   = sOld[s];
        const int n   = sNew[s];
        const int len = sLen[s];
        for (int j = lane; j < len; j += WAVE) {
            // Streamed exactly once: non-temporal hints (CDNA5 TH=NT) keep
            // the near caches clean for the offset/index traffic.
            float v = __builtin_nontemporal_load(src + o + j);
            __builtin_nontemporal_store(v, out + n + j);
        }
    }
}

extern "C" void kernel_launch(void* const* d_in, const int* in_sizes, int n_in,
                              void* d_out, int out_size, void* d_ws, size_t ws_size,
                              hipStream_t stream) {
    (void)n_in; (void)out_size; (void)ws_size;
    const float* src = (const float*)d_in[0];   // tgt_cache_loc  (float32)
    const int*   al  = (const int*)d_in[1];     // accept_length  (int32)
    const int*   f   = (const int*)d_in[2];     // accept_length_filter (int32)
    float*       out = (float*)d_out;

    const int B  = in_sizes[1];
    const int NB = (B + ELEMS_PER_BLOCK - 1) / ELEMS_PER_BLOCK;

    int* blkA = (int*)d_ws;        // NB ints
    int* blkF = blkA + NB;         // NB ints  (~8KB total, fits d_ws)

    k_partial_sums<<<NB, THREADS, 0, stream>>>(al, f, B, blkA, blkF);
    k_scan_sums   <<<1, 1024, 0, stream>>>(blkA, blkF, NB);
    k_scan_copy   <<<NB, THREADS, 0, stream>>>(src, al, f, B, blkA, blkF, out);
}